// GNNLinkPredictor_90452011254251
// MI455X (gfx1250) — compile-verified
//
#include <hip/hip_runtime.h>
#include <hip/hip_bf16.h>

typedef __attribute__((ext_vector_type(2))) float v2f;
typedef __attribute__((ext_vector_type(8))) float v8f;

#define IN_CH 128
#define HIDF  64

// ---------------- scratch zeroing (grid-stride) ----------------
__global__ void zero_f32(float* __restrict__ p, long n) {
  long i = (long)blockIdx.x * blockDim.x + threadIdx.x;
  long stride = (long)gridDim.x * blockDim.x;
  for (; i < n; i += stride) p[i] = 0.0f;
}

// ---------------- degree counts: cnt[dst] += 1 ----------------
__global__ void degree_kernel(const int* __restrict__ edst, float* __restrict__ cnt, int E) {
  int e = blockIdx.x * blockDim.x + threadIdx.x;
  if (e < E)
    (void)__hip_atomic_fetch_add(&cnt[edst[e]], 1.0f, __ATOMIC_RELAXED, __HIP_MEMORY_SCOPE_AGENT);
}

// ---------------- layer-1 scatter: agg[dst, 0:128] += x[src, 0:128] ----------------
// one wave per edge; each lane handles 4 consecutive features (32*4 = 128)
__global__ void scatter128(const int* __restrict__ esrc, const int* __restrict__ edst,
                           const float* __restrict__ x, float* __restrict__ agg, int E) {
  long t = (long)blockIdx.x * blockDim.x + threadIdx.x;
  int e = (int)(t >> 5);
  int lane = (int)(t & 31);
  if (e >= E) return;
  long s = esrc[e];
  long d = edst[e];
  const float4 v = *(const float4*)(x + s * IN_CH + lane * 4);
  float* o = agg + d * IN_CH + lane * 4;
  (void)__hip_atomic_fetch_add(o + 0, v.x, __ATOMIC_RELAXED, __HIP_MEMORY_SCOPE_AGENT);
  (void)__hip_atomic_fetch_add(o + 1, v.y, __ATOMIC_RELAXED, __HIP_MEMORY_SCOPE_AGENT);
  (void)__hip_atomic_fetch_add(o + 2, v.z, __ATOMIC_RELAXED, __HIP_MEMORY_SCOPE_AGENT);
  (void)__hip_atomic_fetch_add(o + 3, v.w, __ATOMIC_RELAXED, __HIP_MEMORY_SCOPE_AGENT);
}

// ---------------- layer-2 scatter: agg[dst, 0:64] += h[src, 0:64] ----------------
__global__ void scatter64(const int* __restrict__ esrc, const int* __restrict__ edst,
                          const float* __restrict__ h, float* __restrict__ agg, int E) {
  long t = (long)blockIdx.x * blockDim.x + threadIdx.x;
  int e = (int)(t >> 5);
  int lane = (int)(t & 31);
  if (e >= E) return;
  long s = esrc[e];
  long d = edst[e];
  const float2 v = *(const float2*)(h + s * HIDF + lane * 2);
  float* o = agg + d * HIDF + lane * 2;
  (void)__hip_atomic_fetch_add(o + 0, v.x, __ATOMIC_RELAXED, __HIP_MEMORY_SCOPE_AGENT);
  (void)__hip_atomic_fetch_add(o + 1, v.y, __ATOMIC_RELAXED, __HIP_MEMORY_SCOPE_AGENT);
}

// ---------------- fused SAGE layer GEMM (fp32 WMMA) ----------------
// out[i, :] = relu( (aggSum[i,:]/max(cnt[i],1)) @ Wl^T + bias + root[i,:] @ Wr^T )
// Wl/Wr are [HIDF, K] row-major, so B = W^T, i.e. B[k][n] = W[n*K + k].
//
// V_WMMA_F32_16X16X4_F32 fragments (wave32):
//   A (16x4, MxK): lanes 0-15 -> (M=m0+lane, K=k+0/k+1); lanes 16-31 -> (K=k+2/k+3)
//   B (4x16, KxN): same K-split per half-wave, N = n0 + (lane&15)
//   C/D (16x16):  VGPR r -> rows m0+r (lanes 0-15) and m0+r+8 (lanes 16-31)
//
// Each wave owns 16 rows x ALL 64 output columns (4 accumulators): the A fragment
// is loaded once per k-step and fed to 4 back-to-back WMMAs, cutting redundant
// A traffic 4x vs a one-tile-per-wave mapping. Block = 8 waves = 128 rows.
__global__ void sage_gemm_wmma(const float* __restrict__ aggSum,
                               const float* __restrict__ root,
                               const float* __restrict__ Wl,
                               const float* __restrict__ Wr,
                               const float* __restrict__ bias,
                               const float* __restrict__ cnt,
                               float* __restrict__ outp,
                               int N, int K) {
  const int lane    = threadIdx.x & 31;
  const int wave    = threadIdx.x >> 5;
  const int lane_lo = lane & 15;
  const int sel     = lane >> 4;                 // 0 -> K+0/1, 1 -> K+2/3
  const int m0      = blockIdx.x * 128 + wave * 16;

  int row = m0 + lane_lo;
  if (row > N - 1) row = N - 1;                  // clamp reads; EXEC stays all-ones
  const float rc = 1.0f / fmaxf(cnt[row], 1.0f);

  v8f acc0 = {}, acc1 = {}, acc2 = {}, acc3 = {};

  // neighbor path: (aggSum * rc) @ Wl^T
  {
    const float* aRow = aggSum + (long)row * K + sel * 2;
    const float* bRow = Wl + (long)lane_lo * K + sel * 2;   // n-tile t at +16*t*K
#pragma unroll 4
    for (int k = 0; k < K; k += 4) {
      v2f a = *(const v2f*)(aRow + k);
      a *= rc;
      v2f w0 = *(const v2f*)(bRow + k);
      v2f w1 = *(const v2f*)(bRow + 16L * K + k);
      v2f w2 = *(const v2f*)(bRow + 32L * K + k);
      v2f w3 = *(const v2f*)(bRow + 48L * K + k);
      acc0 = __builtin_amdgcn_wmma_f32_16x16x4_f32(false, a, false, w0, (short)0, acc0, false, false);
      acc1 = __builtin_amdgcn_wmma_f32_16x16x4_f32(false, a, false, w1, (short)0, acc1, false, false);
      acc2 = __builtin_amdgcn_wmma_f32_16x16x4_f32(false, a, false, w2, (short)0, acc2, false, false);
      acc3 = __builtin_amdgcn_wmma_f32_16x16x4_f32(false, a, false, w3, (short)0, acc3, false, false);
    }
  }
  // root path: root @ Wr^T
  {
    const float* aRow = root + (long)row * K + sel * 2;
    const float* bRow = Wr + (long)lane_lo * K + sel * 2;
#pragma unroll 4
    for (int k = 0; k < K; k += 4) {
      v2f a = *(const v2f*)(aRow + k);
      v2f w0 = *(const v2f*)(bRow + k);
      v2f w1 = *(const v2f*)(bRow + 16L * K + k);
      v2f w2 = *(const v2f*)(bRow + 32L * K + k);
      v2f w3 = *(const v2f*)(bRow + 48L * K + k);
      acc0 = __builtin_amdgcn_wmma_f32_16x16x4_f32(false, a, false, w0, (short)0, acc0, false, false);
      acc1 = __builtin_amdgcn_wmma_f32_16x16x4_f32(false, a, false, w1, (short)0, acc1, false, false);
      acc2 = __builtin_amdgcn_wmma_f32_16x16x4_f32(false, a, false, w2, (short)0, acc2, false, false);
      acc3 = __builtin_amdgcn_wmma_f32_16x16x4_f32(false, a, false, w3, (short)0, acc3, false, false);
    }
  }

  const float b0 = bias[lane_lo];
  const float b1 = bias[16 + lane_lo];
  const float b2 = bias[32 + lane_lo];
  const float b3 = bias[48 + lane_lo];
#pragma unroll
  for (int r = 0; r < 8; ++r) {
    int rowOut = m0 + r + sel * 8;
    if (rowOut < N) {
      float* o = outp + (long)rowOut * HIDF + lane_lo;
      o[0]  = fmaxf(acc0[r] + b0, 0.0f);
      o[16] = fmaxf(acc1[r] + b1, 0.0f);
      o[32] = fmaxf(acc2[r] + b2, 0.0f);
      o[48] = fmaxf(acc3[r] + b3, 0.0f);
    }
  }
}

// ---------------- link prediction: sigmoid(dot(z[src], z[dst])) ----------------
// one wave per query; each lane covers 2 of the 64 features; shuffle reduce
__global__ void link_pred(const float* __restrict__ z, const int* __restrict__ src,
                          const int* __restrict__ dst, float* __restrict__ out, int Q) {
  long t = (long)blockIdx.x * blockDim.x + threadIdx.x;
  int q = (int)(t >> 5);
  int lane = (int)(t & 31);
  if (q >= Q) return;
  long s = src[q];
  long d = dst[q];
  const float* zs = z + s * HIDF;
  const float* zd = z + d * HIDF;
  float p = zs[lane] * zd[lane] + zs[lane + 32] * zd[lane + 32];
#pragma unroll
  for (int off = 16; off >= 1; off >>= 1) p += __shfl_xor(p, off, 32);
  if (lane == 0) out[q] = 1.0f / (1.0f + expf(-p));
}

extern "C" void kernel_launch(void* const* d_in, const int* in_sizes, int n_in,
                              void* d_out, int out_size, void* d_ws, size_t ws_size,
                              hipStream_t stream) {
  const float* x    = (const float*)d_in[0];
  const int*   eidx = (const int*)d_in[1];
  const int*   qsrc = (const int*)d_in[2];
  const int*   qdst = (const int*)d_in[3];
  const float* Wl1  = (const float*)d_in[4];
  const float* bl1  = (const float*)d_in[5];
  const float* Wr1  = (const float*)d_in[6];
  const float* Wl2  = (const float*)d_in[7];
  const float* bl2  = (const float*)d_in[8];
  const float* Wr2  = (const float*)d_in[9];

  const int N = in_sizes[0] / IN_CH;   // 100000
  const int E = in_sizes[1] / 2;       // 1.6M
  const int Q = in_sizes[2];           // 200000
  const int* esrc = eidx;
  const int* edst = eidx + E;

  // workspace layout (floats):
  //   cnt  : Npad              (degree counts)
  //   agg  : N*128             (layer-1 mean numerator; first N*64 reused as layer-2 agg)
  //   h1   : N*64              (layer-1 output)
  //   zbuf : N*64              (layer-2 output)
  float* ws = (float*)d_ws;
  const long Npad = ((long)N + 63) & ~63L;
  float* cnt  = ws;
  float* agg  = ws + Npad;
  float* h1   = agg + (long)N * IN_CH;
  float* zbuf = h1 + (long)N * HIDF;
  float* agg2 = agg;  // alias: agg dead after layer-1 GEMM

  const unsigned scatterBlocks = (unsigned)(((long)E * 32 + 255) / 256);
  const unsigned gemmBlocks    = (unsigned)((N + 127) / 128);

  // layer 1
  zero_f32<<<2048, 256, 0, stream>>>(cnt, Npad + (long)N * IN_CH);
  degree_kernel<<<(E + 255) / 256, 256, 0, stream>>>(edst, cnt, E);
  scatter128<<<scatterBlocks, 256, 0, stream>>>(esrc, edst, x, agg, E);
  sage_gemm_wmma<<<gemmBlocks, 256, 0, stream>>>(agg, x, Wl1, Wr1, bl1, cnt, h1, N, IN_CH);

  // layer 2
  zero_f32<<<2048, 256, 0, stream>>>(agg2, (long)N * HIDF);
  scatter64<<<scatterBlocks, 256, 0, stream>>>(esrc, edst, h1, agg2, E);
  sage_gemm_wmma<<<gemmBlocks, 256, 0, stream>>>(agg2, h1, Wl2, Wr2, bl2, cnt, zbuf, N, HIDF);

  // queries
  link_pred<<<(unsigned)(((long)Q * 32 + 255) / 256), 256, 0, stream>>>(
      zbuf, qsrc, qdst, (float*)d_out, Q);
}